// SeedClusterPrecisionRecall_12618613916166
// MI455X (gfx1250) — compile-verified
//
#include <hip/hip_runtime.h>

typedef float v2f __attribute__((ext_vector_type(2)));
typedef float v8f __attribute__((ext_vector_type(8)));

#define QN 1024
#define CN 256
#define BN 4

// ---------------------------------------------------------------------------
// Kernel 1: row-normalize signatures (one wave32 per row of 256 floats)
// ---------------------------------------------------------------------------
__global__ void scpr_normalize(const float* __restrict__ sig, float* __restrict__ nbuf) {
    int gtid = blockIdx.x * blockDim.x + threadIdx.x;
    int row  = gtid >> 5;
    int lane = gtid & 31;
    if (row >= BN * QN) return;
    const float* rp = sig + (size_t)row * CN;
    float v[8];
    float ss = 0.f;
    #pragma unroll
    for (int t = 0; t < 8; ++t) { v[t] = rp[lane + 32 * t]; ss += v[t] * v[t]; }
    #pragma unroll
    for (int m = 16; m >= 1; m >>= 1) ss += __shfl_xor(ss, m, 32);
    float inv = 1.0f / fmaxf(sqrtf(ss), 1e-8f);
    float* op = nbuf + (size_t)row * CN;
    #pragma unroll
    for (int t = 0; t < 8; ++t) op[lane + 32 * t] = v[t] * inv;
}

// ---------------------------------------------------------------------------
// Kernel 2: fp32 WMMA sim tiles -> thresholded bit-adjacency.
// One wave computes a 16x32 strip of sim[b] via V_WMMA_F32_16X16X4_F32,
// ballots each row against the 3 dup thresholds, stores packed 32-bit words.
// adj layout: adj[((d*BN + b)*QN + row)*32 + wordIdx], wordIdx in [0,32).
// ---------------------------------------------------------------------------
__global__ void __launch_bounds__(128) scpr_simadj(const float* __restrict__ nbuf,
                                                   unsigned* __restrict__ adj) {
    int lane = threadIdx.x & 31;
    int w    = (blockIdx.x * blockDim.x + threadIdx.x) >> 5;  // global wave id
    int ti  = w & 63;          // 64 row tiles of 16
    int tjp = (w >> 6) & 31;   // 32 column strips of 32
    int b   = w >> 11;         // batch

    const float* base = nbuf + (size_t)b * QN * CN;
    int rowA  = ti * 16 + (lane & 15);
    int rowB0 = tjp * 32 + (lane & 15);
    int ksub  = 2 * (lane >> 4);
    const float* pa  = base + (size_t)rowA * CN + ksub;
    const float* pb0 = base + (size_t)rowB0 * CN + ksub;
    const float* pb1 = base + (size_t)(rowB0 + 16) * CN + ksub;

    v8f c0 = {};
    v8f c1 = {};
    #pragma unroll 4
    for (int k = 0; k < CN; k += 4) {
        v2f a  = *(const v2f*)(pa + k);
        v2f b0 = *(const v2f*)(pb0 + k);
        v2f b1 = *(const v2f*)(pb1 + k);
        c0 = __builtin_amdgcn_wmma_f32_16x16x4_f32(false, a, false, b0, (short)0, c0, false, false);
        c1 = __builtin_amdgcn_wmma_f32_16x16x4_f32(false, a, false, b1, (short)0, c1, false, false);
    }

    const float dth[3] = {0.1f, 0.2f, 0.3f};
    #pragma unroll
    for (int d = 0; d < 3; ++d) {
        unsigned* arow = adj + (((size_t)d * BN + b) * QN) * 32;
        #pragma unroll
        for (int v = 0; v < 8; ++v) {
            // ballot bits: [15:0] -> row ti*16+v, cols 0..15 of strip
            //              [31:16]-> row ti*16+v+8, cols 0..15 of strip
            unsigned m0 = __builtin_amdgcn_ballot_w32(c0[v] >= dth[d]);
            unsigned m1 = __builtin_amdgcn_ballot_w32(c1[v] >= dth[d]);
            if (lane == 0) {
                arow[(ti * 16 + v) * 32 + tjp]     = (m0 & 0xFFFFu) | (m1 << 16);
                arow[(ti * 16 + v + 8) * 32 + tjp] = (m0 >> 16) | (m1 & 0xFFFF0000u);
            }
        }
    }
}

// ---------------------------------------------------------------------------
// Kernel 3: per (b,s,d) combo: min-label connected components + TP/FP.
// One 1024-thread workgroup per combo; sel-masked adjacency staged to LDS
// (transposed [word][row] layout -> conflict-free), iterate to fixpoint.
// combo c = b*9 + s*3 + d  (matches output flat order of tp[B,S,D]).
// ---------------------------------------------------------------------------
__global__ void __launch_bounds__(1024) scpr_cluster(const float* __restrict__ logits,
                                                     const unsigned char* __restrict__ matched,
                                                     const unsigned* __restrict__ adjBase,
                                                     float* __restrict__ tpArr,
                                                     float* __restrict__ fpArr) {
    __shared__ unsigned adjLds[32 * QN];     // 128 KB, [word][row]
    __shared__ int labels[QN];
    __shared__ unsigned selw[32];
    __shared__ int changed;
    __shared__ unsigned char comp[QN];
    __shared__ int stp, sfp;

    int c = blockIdx.x;
    int b = c / 9;
    int sd = c % 9;
    int s = sd / 3;
    int d = sd % 3;

    int i    = threadIdx.x;
    int lane = i & 31;
    int wv   = i >> 5;

    // sigmoid(x) >= th  <=>  x >= logit(th); th = {0.3, 0.5, 0.7}
    const float sth[3] = {-0.8472978603872034f, 0.0f, 0.8472978603872034f};
    float lg = logits[b * QN + i];
    bool sel_i = (lg >= sth[s]);
    unsigned sm = __builtin_amdgcn_ballot_w32(sel_i);
    if (lane == 0) selw[wv] = sm;
    labels[i] = sel_i ? i : QN;
    if (i == 0) { stp = 0; sfp = 0; }
    __syncthreads();

    const unsigned* arow = adjBase + (((size_t)d * BN + b) * QN + i) * 32;
    #pragma unroll 4
    for (int w2 = 0; w2 < 32; ++w2)
        adjLds[w2 * QN + i] = arow[w2] & selw[w2];   // mask columns by sel
    __syncthreads();

    // min-label propagation with pointer jumping
    for (;;) {
        if (i == 0) changed = 0;
        __syncthreads();
        if (sel_i) {
            int l0 = labels[i];
            int l  = l0;
            for (int w2 = 0; w2 < 32; ++w2) {
                unsigned bits = adjLds[w2 * QN + i];
                while (bits) {
                    int j = (w2 << 5) + __builtin_ctz(bits);
                    bits &= bits - 1;
                    int lj = labels[j];
                    if (lj < l) l = lj;
                }
            }
            while (l < QN) {                 // shortcut chase (monotone-safe)
                int p = labels[l];
                if (p < l) l = p; else break;
            }
            if (l < l0) { labels[i] = l; changed = 1; }
        }
        __syncthreads();
        if (!changed) break;
        __syncthreads();
    }

    comp[i] = 0;
    __syncthreads();
    if (sel_i && matched[b * QN + i]) comp[labels[i]] = 1;
    __syncthreads();
    if (sel_i && labels[i] == i) {
        if (comp[i]) atomicAdd(&stp, 1); else atomicAdd(&sfp, 1);
    }
    __syncthreads();
    if (i == 0) { tpArr[c] = (float)stp; fpArr[c] = (float)sfp; }
}

// ---------------------------------------------------------------------------
// Kernel 4: derive all 225 outputs.
// out: tp[36] fp[36] fn[36] prec[36] rec[36] ttp[9] tfp[9] tfn[9] mp[9] mr[9]
// ---------------------------------------------------------------------------
__global__ void scpr_finalize(const unsigned char* __restrict__ matched,
                              const float* __restrict__ tpArr,
                              const float* __restrict__ fpArr,
                              float* __restrict__ out) {
    __shared__ float tm[BN];
    int tid = threadIdx.x;
    if (tid < BN) tm[tid] = 0.f;
    __syncthreads();
    for (int idx = tid; idx < BN * QN; idx += blockDim.x)
        if (matched[idx]) atomicAdd(&tm[idx >> 10], 1.0f);
    __syncthreads();

    if (tid < 36) {
        int b = tid / 9;
        float tp = tpArr[tid];
        float fp = fpArr[tid];
        float fn = tm[b] - tp;
        out[tid]       = tp;
        out[36 + tid]  = fp;
        out[72 + tid]  = fn;
        out[108 + tid] = tp / fmaxf(tp + fp, 1.0f);
        out[144 + tid] = tp / fmaxf(tp + fn, 1.0f);
    }
    if (tid < 9) {
        float ttp = 0.f, tfp = 0.f, tfn = 0.f;
        for (int b = 0; b < BN; ++b) {
            int idx = b * 9 + tid;
            float tp = tpArr[idx];
            ttp += tp;
            tfp += fpArr[idx];
            tfn += tm[b] - tp;
        }
        out[180 + tid] = ttp;
        out[189 + tid] = tfp;
        out[198 + tid] = tfn;
        out[207 + tid] = ttp / fmaxf(ttp + tfp, 1.0f);
        out[216 + tid] = ttp / fmaxf(ttp + tfn, 1.0f);
    }
}

// ---------------------------------------------------------------------------
extern "C" void kernel_launch(void* const* d_in, const int* in_sizes, int n_in,
                              void* d_out, int out_size, void* d_ws, size_t ws_size,
                              hipStream_t stream) {
    const float*         sig     = (const float*)d_in[0];          // (4,1024,256) f32
    const float*         logits  = (const float*)d_in[1];          // (4,1024) f32
    const unsigned char* matched = (const unsigned char*)d_in[2];  // (4,1024) bool
    // d_in[3] (matched_gt_indices) is unused by the reference math.
    float* out = (float*)d_out;

    // workspace layout
    float*    nbuf  = (float*)d_ws;                                        // 4 MB
    unsigned* adj   = (unsigned*)((char*)d_ws + (size_t)(4u << 20));       // 1.5 MB
    float*    tpArr = (float*)((char*)d_ws + (size_t)(4u << 20) + 1572864);
    float*    fpArr = tpArr + 64;

    // 1) normalize: 4096 rows, one wave each -> 512 blocks of 256
    scpr_normalize<<<512, 256, 0, stream>>>(sig, nbuf);
    // 2) fp32 WMMA sim + thresholded bit-adjacency: 8192 waves / 4 per block
    scpr_simadj<<<2048, 128, 0, stream>>>(nbuf, adj);
    // 3) connected components + per-combo TP/FP: 36 combos
    scpr_cluster<<<36, 1024, 0, stream>>>(logits, matched, adj, tpArr, fpArr);
    // 4) final 225 outputs
    scpr_finalize<<<1, 256, 0, stream>>>(matched, tpArr, fpArr, out);
}